// EmbeddingLayer_14121852469471
// MI455X (gfx1250) — compile-verified
//
#include <hip/hip_runtime.h>

// Embedding gather + per-field Linear(1,D), fused.
// B=64, L=512, C=12 categorical fields, F=4 continuous fields, D=128.
// Output [B*L, 16, 128] fp32.

typedef __attribute__((ext_vector_type(4))) float v4f;

#define NC    12      // categorical fields
#define NF    4       // continuous fields
#define NROWS 16      // NC + NF
#define DD    128     // embedding dim

// cumsum([0] + VOCABS[:-1]) for VOCABS = [100000,800,200,5000,60,30000,10,8,50,2000,20,16]
__device__ __constant__ long long g_offsets[NC] = {
    0LL, 100000LL, 100800LL, 101000LL, 106000LL, 106060LL,
    136060LL, 136070LL, 136078LL, 136128LL, 138128LL, 138148LL
};

__global__ __launch_bounds__(512)
void emb_fused_kernel(const int*   __restrict__ cat,    // [n_tok, NC]
                      const float* __restrict__ cont,   // [n_tok, NF]
                      const float* __restrict__ table,  // [V_total, DD]
                      const float* __restrict__ wc,     // [NF, DD]
                      const float* __restrict__ bc,     // [NF, DD]
                      float*       __restrict__ out,    // [n_tok, NROWS, DD]
                      int n_tok)
{
    // Per-token scalar staging area: 12 cat indices + 4 cont values (16 x 4B).
    __shared__ __align__(16) unsigned s_stage[NROWS];

    const int tok = blockIdx.x;
    const int tid = threadIdx.x;
    if (tok >= n_tok) return;  // uniform per block

    // --- Stage the 16 per-token scalars into LDS via CDNA5 async copy ---
    if (tid < NROWS) {
        const void* gsrc = (tid < NC)
            ? (const void*)(cat  + (size_t)tok * NC + tid)
            : (const void*)(cont + (size_t)tok * NF + (tid - NC));
        // Low 32 bits of the generic pointer to a __shared__ object are the
        // LDS byte offset (flat->LDS aperture mapping truncates to addr[31:0]).
        unsigned lds_addr = (unsigned)(uintptr_t)(&s_stage[tid]);
        asm volatile("global_load_async_to_lds_b32 %0, %1, off"
                     :: "v"(lds_addr), "v"(gsrc)
                     : "memory");
    }
#if __has_builtin(__builtin_amdgcn_s_wait_asynccnt)
    __builtin_amdgcn_s_wait_asynccnt(0);
#else
    asm volatile("s_wait_asynccnt 0" ::: "memory");
#endif
    __syncthreads();

    // --- One wave32 per output row: lane handles 4 contiguous floats (b128) ---
    const int r    = tid >> 5;        // row within token: 0..15
    const int lane = tid & 31;        // lane within wave
    const int col  = lane << 2;       // float column: 0,4,...,124

    v4f v;
    if (r < NC) {
        const int ci = (int)s_stage[r];
        if (ci == 0) {
            // padding_idx semantics: the pad row of every field sub-table is zero
            v = (v4f)0.0f;
        } else {
            const long long row = (long long)ci + g_offsets[r];
            v = *(const v4f*)(table + row * (long long)DD + col);
        }
    } else {
        const int   f = r - NC;
        const float x = __uint_as_float(s_stage[r]);
        const v4f  wv = *(const v4f*)(wc + f * DD + col);
        const v4f  bv = *(const v4f*)(bc + f * DD + col);
        v = wv * x + bv;              // v_fma per component
    }

    // Streamed 256 MB output: non-temporal store keeps the embedding table
    // resident in the 192 MB L2 instead of being thrashed by the output.
    v4f* dst = (v4f*)(out + (size_t)tok * (NROWS * DD) + r * DD + col);
    __builtin_nontemporal_store(v, dst);
}

extern "C" void kernel_launch(void* const* d_in, const int* in_sizes, int n_in,
                              void* d_out, int out_size, void* d_ws, size_t ws_size,
                              hipStream_t stream)
{
    const int*   cat   = (const int*)  d_in[0];  // [B,L,C] int32
    const float* cont  = (const float*)d_in[1];  // [B,L,F] fp32
    const float* table = (const float*)d_in[2];  // [V_total,D] fp32
    const float* wc    = (const float*)d_in[3];  // [F,D] fp32
    const float* bc    = (const float*)d_in[4];  // [F,D] fp32
    float* out = (float*)d_out;

    const int n_tok = in_sizes[0] / NC;          // B*L = 32768

    emb_fused_kernel<<<dim3(n_tok), dim3(512), 0, stream>>>(
        cat, cont, table, wc, bc, out, n_tok);
}